// GraphModule_59012850647689
// MI455X (gfx1250) — compile-verified
//
#include <hip/hip_runtime.h>
#include <hip/hip_bf16.h>

// 5-layer GCN forward for MI455X (gfx1250, wave32).
// GEMM uses V_WMMA_F32_16X16X4_F32 (fp32 in/out -> matches reference precision).
// Edge aggregation uses hardware global_atomic_add_f32; h/agg (51 MB each) are
// L2-resident on the 192 MB L2, so the random gather/scatter stays on-chip.

#define D 128
#define EPS 1e-5f

typedef __attribute__((ext_vector_type(2))) float v2f;
typedef __attribute__((ext_vector_type(8))) float v8f;

// ---------------------------------------------------------------------------
// GEMM: h = x * W^T  (x: NxD, W: DxD row-major, h[n,j] = dot(x[n,:], W[j,:]))
// Also writes agg = h (self-loop message), folded into the epilogue.
// Block: 256 threads = 8 waves; block covers 128 rows x 128 cols.
// Wave w: 16-row M-strip, 8 N-tiles of 16; K-loop in steps of 4.
// A frag (16x4 f32): lane L (half=L/16, ln=L%16) holds x[m0+ln][kk+half*2+{0,1}]
// B frag (4x16 f32): lane L holds W[n0+ln][kk+half*2+{0,1}]  (B = W^T)
// C/D (16x16 f32):  vgpr v, lane L -> row m0+half*8+v, col n0+ln
// ---------------------------------------------------------------------------
__global__ __launch_bounds__(256) void gcn_gemm_xwT(const float* __restrict__ x,
                                                    const float* __restrict__ W,
                                                    float* __restrict__ h,
                                                    float* __restrict__ agg,
                                                    int N)
{
    const int wave = threadIdx.x >> 5;
    const int lane = threadIdx.x & 31;
    const int half = lane >> 4;
    const int ln   = lane & 15;
    const int m0   = blockIdx.x * 128 + wave * 16;
    if (m0 >= N) return;                 // wave-uniform: EXEC stays all-ones

    int ra = m0 + ln;
    if (ra >= N) ra = N - 1;             // N % 16 == 0 in practice; safe clamp
    const float* xrow = x + (size_t)ra * D;

    v8f acc[8] = {};

    #pragma unroll 4
    for (int kk = 0; kk < D; kk += 4) {
        const int kb = kk + half * 2;
        v2f a = *(const v2f*)(xrow + kb);
        #pragma unroll
        for (int j = 0; j < 8; ++j) {
            v2f bfr = *(const v2f*)(W + (size_t)(j * 16 + ln) * D + kb);
            acc[j] = __builtin_amdgcn_wmma_f32_16x16x4_f32(
                /*neg_a=*/false, a, /*neg_b=*/false, bfr,
                /*c_mod=*/(short)0, acc[j], /*reuse_a=*/false, /*reuse_b=*/false);
        }
    }

    #pragma unroll
    for (int j = 0; j < 8; ++j) {
        const int col = j * 16 + ln;
        #pragma unroll
        for (int v = 0; v < 8; ++v) {
            const int m = m0 + half * 8 + v;
            if (m < N) {
                const size_t idx = (size_t)m * D + col;
                const float val = acc[j][v];
                h[idx]   = val;
                agg[idx] = val;   // self-loop term: agg starts as h
            }
        }
    }
}

// ---------------------------------------------------------------------------
// Degree helpers (computed once; deg = in-degree + 1, then inverted in place)
// ---------------------------------------------------------------------------
__global__ void gcn_init_deg(float* __restrict__ deg, int N)
{
    int t = blockIdx.x * blockDim.x + threadIdx.x;
    if (t < N) deg[t] = 1.0f;
}

__global__ void gcn_count_deg(const int* __restrict__ dst, float* __restrict__ deg, int E)
{
    int t = blockIdx.x * blockDim.x + threadIdx.x;
    if (t < E) unsafeAtomicAdd(&deg[dst[t]], 1.0f);
}

__global__ void gcn_invert_deg(float* __restrict__ deg, int N)
{
    int t = blockIdx.x * blockDim.x + threadIdx.x;
    if (t < N) deg[t] = 1.0f / deg[t];
}

// ---------------------------------------------------------------------------
// Edge scatter: agg[dst[e], :] += h[src[e], :]   (128 threads per edge)
// ---------------------------------------------------------------------------
__global__ __launch_bounds__(256) void gcn_scatter(const float* __restrict__ h,
                                                   const int* __restrict__ src,
                                                   const int* __restrict__ dst,
                                                   float* __restrict__ agg,
                                                   int E)
{
    const int t = blockIdx.x * 256 + threadIdx.x;
    const int e = t >> 7;
    const int f = t & 127;
    if (e < E) {
        const int s = src[e];
        const int d = dst[e];
        unsafeAtomicAdd(&agg[(size_t)d * D + f], h[(size_t)s * D + f]);
    }
}

// ---------------------------------------------------------------------------
// Per-layer stats reset
// ---------------------------------------------------------------------------
__global__ void gcn_zero_stats(float* __restrict__ sum, float* __restrict__ sumsq)
{
    const int t = threadIdx.x;
    if (t < D) sum[t] = 0.0f; else sumsq[t - D] = 0.0f;
}

// ---------------------------------------------------------------------------
// agg = agg * inv_deg + b (in place), accumulating per-column sum / sumsq.
// Block covers 128 rows; thread (f = t&127, half = t>>7) walks rows.
// ---------------------------------------------------------------------------
__global__ __launch_bounds__(256) void gcn_scale_stats(float* __restrict__ agg,
                                                       const float* __restrict__ inv_deg,
                                                       const float* __restrict__ bias,
                                                       float* __restrict__ sum,
                                                       float* __restrict__ sumsq,
                                                       int N)
{
    __shared__ float ls[256];
    __shared__ float lss[256];
    const int f    = threadIdx.x & 127;
    const int half = threadIdx.x >> 7;
    const int row0 = blockIdx.x * 128;
    const float bf = bias[f];

    float s = 0.0f, ss = 0.0f;
    for (int r = half; r < 128; r += 2) {
        const int row = row0 + r;
        if (row < N) {
            const size_t idx = (size_t)row * D + f;
            const float v = agg[idx] * inv_deg[row] + bf;
            agg[idx] = v;
            s  += v;
            ss += v * v;
        }
    }
    ls[threadIdx.x]  = s;
    lss[threadIdx.x] = ss;
    __syncthreads();
    if (half == 0) {
        s  += ls[threadIdx.x + 128];
        ss += lss[threadIdx.x + 128];
        unsafeAtomicAdd(&sum[f],   s);
        unsafeAtomicAdd(&sumsq[f], ss);
    }
}

// ---------------------------------------------------------------------------
// BatchNorm (no affine, biased var) + ReLU:  out = relu((v - mean)*rsqrt(var+eps))
// ---------------------------------------------------------------------------
__global__ __launch_bounds__(256) void gcn_bn_relu(const float* __restrict__ agg,
                                                   const float* __restrict__ sum,
                                                   const float* __restrict__ sumsq,
                                                   float* __restrict__ out,
                                                   int N)
{
    const int t = blockIdx.x * 256 + threadIdx.x;
    const int total = N * D;
    if (t < total) {
        const int f = t & 127;
        const float invN = 1.0f / (float)N;
        const float mean = sum[f] * invN;
        float var = sumsq[f] * invN - mean * mean;
        var = fmaxf(var, 0.0f);
        const float v = (agg[t] - mean) * rsqrtf(var + EPS);
        out[t] = fmaxf(v, 0.0f);
    }
}

// ---------------------------------------------------------------------------
extern "C" void kernel_launch(void* const* d_in, const int* in_sizes, int n_in,
                              void* d_out, int out_size, void* d_ws, size_t ws_size,
                              hipStream_t stream)
{
    const float* x0  = (const float*)d_in[0];
    const int*   ei  = (const int*)d_in[1];
    const int    N   = in_sizes[0] / D;
    const int    E   = in_sizes[1] / 2;
    const int*   src = ei;          // edge_index[0, :]
    const int*   dst = ei + E;      // edge_index[1, :]

    const float* Ws[5];
    const float* bs[5];
    for (int i = 0; i < 5; ++i) {
        Ws[i] = (const float*)d_in[2 + 2 * i];
        bs[i] = (const float*)d_in[3 + 2 * i];
    }
    float* out = (float*)d_out;

    // Workspace carve-up (256B aligned): deg[N], h[N*D], agg[N*D], sum[D], sumsq[D]
    char* ws = (char*)d_ws;
    size_t off = 0;
    auto align256 = [](size_t v) { return (v + 255) & ~(size_t)255; };
    float* deg   = (float*)(ws + off); off = align256(off + (size_t)N * 4);
    float* h     = (float*)(ws + off); off = align256(off + (size_t)N * D * 4);
    float* agg   = (float*)(ws + off); off = align256(off + (size_t)N * D * 4);
    float* sum   = (float*)(ws + off); off = align256(off + (size_t)D * 4);
    float* sumsq = (float*)(ws + off); off = align256(off + (size_t)D * 4);
    (void)ws_size; (void)n_in; (void)out_size;

    // Degree with self-loops, then invert (depends only on edge_index)
    gcn_init_deg  <<<(N + 255) / 256, 256, 0, stream>>>(deg, N);
    gcn_count_deg <<<(E + 255) / 256, 256, 0, stream>>>(dst, deg, E);
    gcn_invert_deg<<<(N + 255) / 256, 256, 0, stream>>>(deg, N);

    const int gemm_blocks = (N + 127) / 128;
    const int scat_blocks = (int)(((size_t)E * D + 255) / 256);
    const int elem_blocks = (int)(((size_t)N * D + 255) / 256);

    const float* xin = x0;
    for (int l = 0; l < 5; ++l) {
        gcn_zero_stats <<<1, 256, 0, stream>>>(sum, sumsq);
        gcn_gemm_xwT   <<<gemm_blocks, 256, 0, stream>>>(xin, Ws[l], h, agg, N);
        gcn_scatter    <<<scat_blocks, 256, 0, stream>>>(h, src, dst, agg, E);
        gcn_scale_stats<<<gemm_blocks, 256, 0, stream>>>(agg, deg, bs[l], sum, sumsq, N);
        gcn_bn_relu    <<<elem_blocks, 256, 0, stream>>>(agg, sum, sumsq, out, N);
        xin = out;   // ping-pong through d_out; final layer result lands in d_out
    }
}